// SLAD_55972013801849
// MI455X (gfx1250) — compile-verified
//
#include <hip/hip_runtime.h>
#include <math.h>

typedef float v2f __attribute__((ext_vector_type(2)));
typedef float v8f __attribute__((ext_vector_type(8)));

#define BB   16384   // batch
#define FF   256     // features (== L, idx is a full permutation)
#define EE   4
#define DD   10
#define UU   16
#define HH   8
#define LLL  256
#define NED  (EE * DD)        // 40 (e,d) pairs == 40 N-tiles of 16
#define NCOL (NED * UU)       // 640 output columns
#define MB   64               // rows (b) per workgroup
#define XSTR 258              // padded LDS row stride for X tile (bank-conflict avoidance)
#define PSTR 642              // padded LDS row stride for proj tile

// ---------------------------------------------------------------------------
// Kernel 1: scatter-permute W -> Wp so the gather-GEMM becomes a plain GEMM.
// Wp[(ed*U+u)*F + idx[ed,l]] = W[(ed*U+u)*L + l]
// ---------------------------------------------------------------------------
__global__ __launch_bounds__(256) void slad_permute_w(const float* __restrict__ W,
                                                      const int* __restrict__ idx,
                                                      float* __restrict__ Wp) {
  int t = blockIdx.x * blockDim.x + threadIdx.x;
  if (t >= NED * UU * LLL) return;
  int l  = t & (LLL - 1);
  int u  = (t >> 8) & (UU - 1);
  int ed = t >> 12;                       // 0..39
  int f  = idx[ed * LLL + l];
  Wp[(ed * UU + u) * FF + f] = W[(ed * UU + u) * LLL + l];
}

// ---------------------------------------------------------------------------
// Kernel 2 (fused): X copy-out + WMMA GEMM (X @ Wp^T) + MLP/leaky-ReLU +
// JS divergence + 4-way loss reduction. One workgroup = 64 rows of X.
// 8 waves; wave w computes N-tiles {w, w+8, ..., w+32}, 4 M-tiles each
// (register blocking: one B fragment feeds 4 WMMAs).
// ---------------------------------------------------------------------------
__global__ __launch_bounds__(256) void slad_main(const float* __restrict__ X,
                                                 const float* __restrict__ Wp,
                                                 const int* __restrict__ nf,
                                                 const float* __restrict__ W1,
                                                 const float* __restrict__ b1,
                                                 const float* __restrict__ W2,
                                                 const float* __restrict__ b2,
                                                 float* __restrict__ Xout,
                                                 float* __restrict__ loss) {
  __shared__ float Xs[MB * XSTR];   // 64 x 256 X tile (padded)
  __shared__ float Ps[MB * PSTR];   // 64 x 640 proj tile (padded)
  __shared__ float Js[EE * MB];     // per-(e, b_local) JS values

  const int tid = threadIdx.x;
  const int b0  = blockIdx.x * MB;

  // ---- Stage X tile into LDS and stream the identity copy to d_out --------
  for (int p = 0; p < MB / 4; ++p) {
    int row = p * 4 + (tid >> 6);
    int c4  = (tid & 63) * 4;
    const float4 v = *(const float4*)&X[(size_t)(b0 + row) * FF + c4];
    *(float4*)&Xs[row * XSTR + c4]                  = v;
    *(float4*)&Xout[(size_t)(b0 + row) * FF + c4]   = v;
  }
  __syncthreads();

  // ---- WMMA GEMM ----------------------------------------------------------
  const int wave = tid >> 5;
  const int lane = tid & 31;
  const int fm   = lane & 15;          // M (for A) / N (for B) within tile
  const int kb   = (lane >> 4) << 1;   // K sub-offset {0,2}

  for (int nti = 0; nti < 5; ++nti) {
    const int nt   = wave + nti * 8;   // 0..39  (== ed pair)
    const int col0 = nt * 16;
    v8f acc[4] = {};
    for (int k0 = 0; k0 < FF; k0 += 4) {
      // B fragment: B[k,n] = Wp[col0+n][k]  (Wp row-contiguous in k)
      v2f bfrag = *(const v2f*)&Wp[(size_t)(col0 + fm) * FF + k0 + kb];
#pragma unroll
      for (int mt = 0; mt < 4; ++mt) {
        v2f afrag = *(const v2f*)&Xs[(mt * 16 + fm) * XSTR + k0 + kb];
        acc[mt] = __builtin_amdgcn_wmma_f32_16x16x4_f32(
            false, afrag, false, bfrag, (short)0, acc[mt], false, false);
      }
    }
    // C/D layout: vgpr r holds row r (lanes 0-15) or r+8 (lanes 16-31), col = lane%16
#pragma unroll
    for (int mt = 0; mt < 4; ++mt)
#pragma unroll
      for (int r = 0; r < 8; ++r) {
        int row = mt * 16 + ((lane < 16) ? r : r + 8);
        Ps[row * PSTR + col0 + fm] = acc[mt][r];
      }
  }
  __syncthreads();

  // ---- Epilogue: MLP + JS divergence, one thread per (e, b_local) ---------
  {
    const int e  = tid >> 6;   // 0..3
    const int bl = tid & 63;   // 0..63
    float pred[DD], yv[DD];
#pragma unroll
    for (int d = 0; d < DD; ++d) {
      const float* xv = &Ps[bl * PSTR + (e * DD + d) * UU];
      float acc2 = b2[0];
#pragma unroll
      for (int j = 0; j < HH; ++j) {
        float a = b1[j];
#pragma unroll
        for (int u = 0; u < UU; ++u) a = fmaf(xv[u], W1[j * UU + u], a);
        a = (a >= 0.0f) ? a : 0.01f * a;           // leaky_relu(., 0.01)
        acc2 = fmaf(a, W2[j], acc2);
      }
      pred[d] = acc2;
      yv[d]   = (float)nf[e * DD + d] * (200.0f / 16.0f);  // nf/U*MAG
    }

    float mp = pred[0], mq = yv[0];
    for (int d = 1; d < DD; ++d) { mp = fmaxf(mp, pred[d]); mq = fmaxf(mq, yv[d]); }
    float p[DD], q[DD], sp = 0.0f, sq = 0.0f;
    for (int d = 0; d < DD; ++d) {
      p[d] = expf(pred[d] - mp); sp += p[d];
      q[d] = expf(yv[d]  - mq); sq += q[d];
    }
    const float isp = 1.0f / sp, isq = 1.0f / sq;
    float mP = -1e30f, mQ = -1e30f;
    for (int d = 0; d < DD; ++d) {
      p[d] = p[d] * isp + 1e-6f; q[d] = q[d] * isq + 1e-6f;
      mP = fmaxf(mP, p[d]); mQ = fmaxf(mQ, q[d]);
    }
    float sP = 0.0f, sQ = 0.0f;
    for (int d = 0; d < DD; ++d) { sP += expf(p[d] - mP); sQ += expf(q[d] - mQ); }
    const float lseP = mP + logf(sP), lseQ = mQ + logf(sQ);
    float jss = 0.0f;
    for (int d = 0; d < DD; ++d) {
      float md = 0.5f * (p[d] + q[d]);
      float lm = logf(md);
      jss += md * (2.0f * lm - (p[d] - lseP) - (q[d] - lseQ));
    }
    Js[e * MB + bl] = jss;
  }
  __syncthreads();

  // ---- loss[i] = mean over 4 consecutive b of js[e][b] --------------------
  if (tid < 64) {
    int e = tid >> 4, g = tid & 15;
    const float* js = &Js[e * MB + g * 4];
    loss[e * (BB / 4) + (b0 >> 2) + g] = 0.25f * (js[0] + js[1] + js[2] + js[3]);
  }
}

// ---------------------------------------------------------------------------
extern "C" void kernel_launch(void* const* d_in, const int* in_sizes, int n_in,
                              void* d_out, int out_size, void* d_ws, size_t ws_size,
                              hipStream_t stream) {
  const float* X   = (const float*)d_in[0];
  const int*   idx = (const int*)d_in[1];
  const int*   nf  = (const int*)d_in[2];
  const float* W   = (const float*)d_in[3];
  const float* W1  = (const float*)d_in[4];
  const float* b1  = (const float*)d_in[5];
  const float* W2  = (const float*)d_in[6];
  const float* b2  = (const float*)d_in[7];

  float* Wp   = (float*)d_ws;                 // 640*256 floats = 640 KB scratch
  float* Xout = (float*)d_out;                // X passthrough (16384*256)
  float* loss = Xout + (size_t)BB * FF;       // then loss (16384)

  slad_permute_w<<<(NED * UU * LLL + 255) / 256, 256, 0, stream>>>(W, idx, Wp);
  slad_main<<<BB / MB, 256, 0, stream>>>(X, Wp, nf, W1, b1, W2, b2, Xout, loss);
}